// HeteroRGCN_74457553043643
// MI455X (gfx1250) — compile-verified
//
#include <hip/hip_runtime.h>

typedef float v2f __attribute__((ext_vector_type(2)));
typedef float v8f __attribute__((ext_vector_type(8)));

#define D      128
#define MTILE  64
#define ROWPAD 130   // LDS row stride in floats (even -> 8B aligned v2f, 130%64=2 avoids bank conflicts)

// Y[r, :] = relu(X[r, :] @ W + b)   for r in [0, nrows)
// Block: 256 threads = 8 waves. Block handles MTILE=64 rows; wave w handles cols [16w, 16w+16)
// with 4 stacked 16x16 accumulators. Uses V_WMMA_F32_16X16X4_F32 (native fp32 tensor op).
__launch_bounds__(256)
__global__ void rgcn_gemm_relu_wmma(const float* __restrict__ X,
                                    const float* __restrict__ W,
                                    const float* __restrict__ bias,
                                    float* __restrict__ Y,
                                    int nrows)
{
    __shared__ float sX[MTILE * ROWPAD];

    const int tid  = threadIdx.x;
    const int wave = tid >> 5;
    const int lane = tid & 31;
    const int r0   = blockIdx.x * MTILE;

    // Cooperative load of X tile (64 x 128 fp32) into LDS, zero-padded past nrows.
    #pragma unroll
    for (int i = 0; i < (MTILE * D / 4) / 256; ++i) {
        const int f    = tid + i * 256;     // float4 index; 32 float4 per row
        const int row  = f >> 5;
        const int c4   = (f & 31) << 2;
        float4 v = make_float4(0.f, 0.f, 0.f, 0.f);
        const int grow = r0 + row;
        if (grow < nrows)
            v = *(const float4*)(X + (size_t)grow * D + c4);
        float* p = &sX[row * ROWPAD + c4];
        p[0] = v.x; p[1] = v.y; p[2] = v.z; p[3] = v.w;
    }
    __syncthreads();

    const int   col  = (wave << 4) + (lane & 15);
    const int   koff = (lane >> 4) << 1;      // lanes 0-15: K+0..1, lanes 16-31: K+2..3
    const int   arow = lane & 15;             // A matrix: M = lane & 15
    const float bv   = bias[col];

    v8f acc[4];
    #pragma unroll
    for (int t = 0; t < 4; ++t) acc[t] = (v8f){0.f,0.f,0.f,0.f,0.f,0.f,0.f,0.f};

    #pragma unroll
    for (int k = 0; k < D; k += 4) {
        v2f b;
        b.x = W[(size_t)(k + koff) * D + col];
        b.y = W[(size_t)(k + koff + 1) * D + col];
        v2f a0 = *(const v2f*)&sX[(0 * 16 + arow) * ROWPAD + k + koff];
        v2f a1 = *(const v2f*)&sX[(1 * 16 + arow) * ROWPAD + k + koff];
        v2f a2 = *(const v2f*)&sX[(2 * 16 + arow) * ROWPAD + k + koff];
        v2f a3 = *(const v2f*)&sX[(3 * 16 + arow) * ROWPAD + k + koff];
        acc[0] = __builtin_amdgcn_wmma_f32_16x16x4_f32(false, a0, false, b, (short)0, acc[0], false, false);
        acc[1] = __builtin_amdgcn_wmma_f32_16x16x4_f32(false, a1, false, b, (short)0, acc[1], false, false);
        acc[2] = __builtin_amdgcn_wmma_f32_16x16x4_f32(false, a2, false, b, (short)0, acc[2], false, false);
        acc[3] = __builtin_amdgcn_wmma_f32_16x16x4_f32(false, a3, false, b, (short)0, acc[3], false, false);
    }

    // D layout: VGPR r -> M = (lane>=16 ? 8 : 0) + r, N = lane & 15
    const int mhi = (lane & 16) ? 8 : 0;
    #pragma unroll
    for (int t = 0; t < 4; ++t) {
        #pragma unroll
        for (int r = 0; r < 8; ++r) {
            const int grow = r0 + t * 16 + mhi + r;
            if (grow < nrows)
                Y[(size_t)grow * D + col] = fmaxf(acc[t][r] + bv, 0.f);
        }
    }
}

// out[dst[e], :] += Y[src[e], :]   — one wave per edge, lane handles 4 columns (float4).
__launch_bounds__(256)
__global__ void rgcn_scatter_add(const float* __restrict__ Y,
                                 const int* __restrict__ ei,   // [2, E] flat: src then dst
                                 int nedges,
                                 float* __restrict__ out)
{
    const int wid  = (int)(((size_t)blockIdx.x * blockDim.x + threadIdx.x) >> 5);
    const int lane = threadIdx.x & 31;
    if (wid >= nedges) return;
    const int s = ei[wid];
    const int d = ei[nedges + wid];
    const float4 v = *(const float4*)(Y + (size_t)s * D + (lane << 2));
    float* o = out + (size_t)d * D + (lane << 2);
    __hip_atomic_fetch_add(o + 0, v.x, __ATOMIC_RELAXED, __HIP_MEMORY_SCOPE_AGENT);
    __hip_atomic_fetch_add(o + 1, v.y, __ATOMIC_RELAXED, __HIP_MEMORY_SCOPE_AGENT);
    __hip_atomic_fetch_add(o + 2, v.z, __ATOMIC_RELAXED, __HIP_MEMORY_SCOPE_AGENT);
    __hip_atomic_fetch_add(o + 3, v.w, __ATOMIC_RELAXED, __HIP_MEMORY_SCOPE_AGENT);
}

extern "C" void kernel_launch(void* const* d_in, const int* in_sizes, int n_in,
                              void* d_out, int out_size, void* d_ws, size_t ws_size,
                              hipStream_t stream) {
    // setup_inputs() order:
    // 0:x_user 1:x_item 2:ei_follows 3:ei_rates
    // 4:W0_follows 5:b0_follows 6:W0_rates 7:b0_rates
    // 8:W1_follows 9:b1_follows 10:W1_rates 11:b1_rates
    const float* x_user = (const float*)d_in[0];
    const int*   ei_f   = (const int*)d_in[2];
    const int*   ei_r   = (const int*)d_in[3];
    const float* W0f    = (const float*)d_in[4];
    const float* b0f    = (const float*)d_in[5];
    const float* W1f    = (const float*)d_in[8];
    const float* b1f    = (const float*)d_in[9];
    const float* W1r    = (const float*)d_in[10];
    const float* b1r    = (const float*)d_in[11];

    const long N = in_sizes[0] / D;
    const long E = in_sizes[2] / 2;

    float* out_u = (float*)d_out;         // xu after layer 1
    float* out_i = out_u + (size_t)N * D; // xi after layer 1
    float* bufA  = (float*)d_ws;                 // per-node transformed features (51.2 MB)
    float* bufH  = bufA + (size_t)N * D;         // layer-0 aggregated user features (51.2 MB)

    const dim3 blk(256);
    const dim3 ggrid((unsigned)((N + MTILE - 1) / MTILE));
    const dim3 sgrid((unsigned)((E + 7) / 8));   // 8 waves (edges) per block

    // Layer 0, 'follows' (the layer-0 'rates' conv is dead code in the reference):
    //   h = scatter(relu(x_user @ W0f + b0f))
    hipMemsetAsync(bufH, 0, (size_t)N * D * sizeof(float), stream);
    rgcn_gemm_relu_wmma<<<ggrid, blk, 0, stream>>>(x_user, W0f, b0f, bufA, (int)N);
    rgcn_scatter_add<<<sgrid, blk, 0, stream>>>(bufA, ei_f, (int)E, bufH);

    // Layer 1, 'follows': out_u = scatter(relu(h @ W1f + b1f))
    rgcn_gemm_relu_wmma<<<ggrid, blk, 0, stream>>>(bufH, W1f, b1f, bufA, (int)N);
    hipMemsetAsync(out_u, 0, (size_t)N * D * sizeof(float), stream);
    rgcn_scatter_add<<<sgrid, blk, 0, stream>>>(bufA, ei_f, (int)E, out_u);

    // Layer 1, 'rates': out_i = scatter(relu(h @ W1r + b1r))
    rgcn_gemm_relu_wmma<<<ggrid, blk, 0, stream>>>(bufH, W1r, b1r, bufA, (int)N);
    hipMemsetAsync(out_i, 0, (size_t)N * D * sizeof(float), stream);
    rgcn_scatter_add<<<sgrid, blk, 0, stream>>>(bufA, ei_r, (int)E, out_i);
}